// ResGCN_44736379355415
// MI455X (gfx1250) — compile-verified
//
#include <hip/hip_runtime.h>
#include <hip/hip_bf16.h>
#include <math.h>

#define N_NODES 100000
#define N_EDGES 1200000
#define C_IN 64
#define H_DIM 64
#define OUT_DIM 10
#define G_BATCH 128
#define L_LAYERS 3
#define BN_EPS 1e-5f

typedef float v2f __attribute__((ext_vector_type(2)));
typedef float v8f __attribute__((ext_vector_type(8)));

// ---------------------------------------------------------------------------
// small utility kernels
// ---------------------------------------------------------------------------
__global__ void zero_f32(float* p, int n) {
    int t = blockIdx.x * blockDim.x + threadIdx.x;
    if (t < n) p[t] = 0.0f;
}

__global__ void fill_f32(float* p, float v, int n) {
    int t = blockIdx.x * blockDim.x + threadIdx.x;
    if (t < n) p[t] = v;
}

// deg[src[e]] += 1 for real edges (deg pre-initialized to 1.0 for self loops)
__global__ void deg_accum(const int* __restrict__ src, float* __restrict__ deg, int e_cnt) {
    int t = blockIdx.x * blockDim.x + threadIdx.x;
    if (t < e_cnt) atomicAdd(&deg[src[t]], 1.0f);
}

// dis = rsqrt(deg) in place (deg >= 1 always, self loops)
__global__ void deg_to_dis(float* __restrict__ d, int n) {
    int t = blockIdx.x * blockDim.x + threadIdx.x;
    if (t < n) d[t] = rsqrtf(d[t]);
}

__global__ void norm_kernel(const int* __restrict__ src, const int* __restrict__ dst,
                            const float* __restrict__ dis, float* __restrict__ norm, int e_cnt) {
    int t = blockIdx.x * blockDim.x + threadIdx.x;
    if (t < e_cnt) norm[t] = dis[src[t]] * dis[dst[t]];
}

// ---------------------------------------------------------------------------
// BatchNorm statistics over rows of a [nrows, 64] matrix
// block: 256 threads = 64 channels x 4 row-lanes; atomicAdd partials
// ---------------------------------------------------------------------------
__global__ void bn_stats(const float* __restrict__ X, float* __restrict__ gsum,
                         float* __restrict__ gsumsq, int nrows) {
    int c = threadIdx.x & 63;
    int r = threadIdx.x >> 6;  // 0..3
    float s = 0.0f, q = 0.0f;
    for (int i = blockIdx.x * 4 + r; i < nrows; i += gridDim.x * 4) {
        float v = X[(size_t)i * 64 + c];
        s += v;
        q += v * v;
    }
    __shared__ float ls[4][64];
    __shared__ float lq[4][64];
    ls[r][c] = s;
    lq[r][c] = q;
    __syncthreads();
    if (threadIdx.x < 64) {
        float ts = ls[0][c] + ls[1][c] + ls[2][c] + ls[3][c];
        float tq = lq[0][c] + lq[1][c] + lq[2][c] + lq[3][c];
        atomicAdd(&gsum[c], ts);
        atomicAdd(&gsumsq[c], tq);
    }
}

// fold BN into per-channel scale/shift:  bn(x) = x*scale + shift
__global__ void bn_finalize(const float* __restrict__ gsum, const float* __restrict__ gsumsq,
                            const float* __restrict__ gamma, const float* __restrict__ beta,
                            float* __restrict__ scale, float* __restrict__ shift, int nrows) {
    int c = threadIdx.x;
    if (c < 64) {
        float inv_n = 1.0f / (float)nrows;
        float mu = gsum[c] * inv_n;
        float var = gsumsq[c] * inv_n - mu * mu;
        float sc = gamma[c] * rsqrtf(var + BN_EPS);
        scale[c] = sc;
        shift[c] = beta[c] - mu * sc;
    }
}

// ---------------------------------------------------------------------------
// WMMA GEMM:  Y[nrows,64] = (X*scale + shift) @ W  (+bias, relu if mode==1)
// one wave -> 16 rows x 64 cols; K=64 in steps of 4 via v_wmma_f32_16x16x4_f32
// nrows must be a multiple of 16 (N=100000 = 6250*16)
// ---------------------------------------------------------------------------
__global__ void gemm_bn_wmma(const float* __restrict__ X, const float* __restrict__ W,
                             const float* __restrict__ scale, const float* __restrict__ shift,
                             const float* __restrict__ bias, float* __restrict__ Y,
                             int nrows, int mode) {
    int wave = blockIdx.x * (blockDim.x >> 5) + (threadIdx.x >> 5);
    int lane = threadIdx.x & 31;
    int row0 = wave * 16;
    if (row0 >= nrows) return;  // wave-uniform: EXEC stays all-ones for WMMA

    int m = lane & 15;           // A row / B,C column within tile
    int khalf = (lane >> 4) * 2; // lanes 16-31 hold K+2,K+3

    v8f acc0 = {}, acc1 = {}, acc2 = {}, acc3 = {};
    const float* xr = X + (size_t)(row0 + m) * 64;

    for (int k0 = 0; k0 < 64; k0 += 4) {
        int ka = k0 + khalf;
        // A fragment with BatchNorm affine folded in
        v2f a;
        a.x = xr[ka] * scale[ka] + shift[ka];
        a.y = xr[ka + 1] * scale[ka + 1] + shift[ka + 1];
        // B fragments for the four 16-column blocks
        const float* w0 = W + (size_t)ka * 64 + m;
        const float* w1 = W + (size_t)(ka + 1) * 64 + m;
        v2f b0; b0.x = w0[0];  b0.y = w1[0];
        v2f b1; b1.x = w0[16]; b1.y = w1[16];
        v2f b2; b2.x = w0[32]; b2.y = w1[32];
        v2f b3; b3.x = w0[48]; b3.y = w1[48];
        acc0 = __builtin_amdgcn_wmma_f32_16x16x4_f32(false, a, false, b0, (short)0, acc0, false, false);
        acc1 = __builtin_amdgcn_wmma_f32_16x16x4_f32(false, a, false, b1, (short)0, acc1, false, false);
        acc2 = __builtin_amdgcn_wmma_f32_16x16x4_f32(false, a, false, b2, (short)0, acc2, false, false);
        acc3 = __builtin_amdgcn_wmma_f32_16x16x4_f32(false, a, false, b3, (short)0, acc3, false, false);
    }

    // write back: C/D layout -> VGPR v: lanes 0-15 row v, lanes 16-31 row v+8
    int col = lane & 15;
    int rhalf = (lane >> 4) * 8;
    v8f accs[4] = {acc0, acc1, acc2, acc3};
#pragma unroll
    for (int nb = 0; nb < 4; nb++) {
#pragma unroll
        for (int v = 0; v < 8; v++) {
            float val = accs[nb][v];
            int r = row0 + v + rhalf;
            int c = nb * 16 + col;
            if (mode) {
                val += bias[c];
                val = fmaxf(val, 0.0f);
            }
            Y[(size_t)r * 64 + c] = val;
        }
    }
}

// ---------------------------------------------------------------------------
// scatter stage: acc[i] starts as self-loop term dis[i]^2 * hw[i]
// ---------------------------------------------------------------------------
__global__ void selfloop_init(const float* __restrict__ hw, const float* __restrict__ dis,
                              float* __restrict__ acc, int n) {
    int t = blockIdx.x * blockDim.x + threadIdx.x;
    int i = t >> 4;
    if (i >= n) return;
    int q = (t & 15) * 4;
    float d = dis[i];
    float w = d * d;
    const float4 v = *(const float4*)(hw + (size_t)i * 64 + q);
    float4 o;
    o.x = w * v.x; o.y = w * v.y; o.z = w * v.z; o.w = w * v.w;
    *(float4*)(acc + (size_t)i * 64 + q) = o;
}

// per edge: acc[dst] += norm[e] * hw[src]; 16 threads/edge, float4 gather
__global__ void edge_scatter(const float* __restrict__ hw, const int* __restrict__ src,
                             const int* __restrict__ dst, const float* __restrict__ norm,
                             float* __restrict__ acc, int e_cnt) {
    int t = blockIdx.x * blockDim.x + threadIdx.x;
    int e = t >> 4;
    if (e >= e_cnt) return;
    int q = (t & 15) * 4;
    int s = src[e];
    int d = dst[e];
    float nm = norm[e];
    const float4 v = *(const float4*)(hw + (size_t)s * 64 + q);
    float* ap = acc + (size_t)d * 64 + q;
    atomicAdd(ap + 0, nm * v.x);
    atomicAdd(ap + 1, nm * v.y);
    atomicAdd(ap + 2, nm * v.z);
    atomicAdd(ap + 3, nm * v.w);
}

__global__ void bias_relu(const float* __restrict__ acc, const float* __restrict__ bias,
                          float* __restrict__ out, int n) {
    int t = blockIdx.x * blockDim.x + threadIdx.x;
    int i = t >> 4;
    if (i >= n) return;
    int q = (t & 15) * 4;
    const float4 v = *(const float4*)(acc + (size_t)i * 64 + q);
    float4 o;
    o.x = fmaxf(v.x + bias[q + 0], 0.0f);
    o.y = fmaxf(v.y + bias[q + 1], 0.0f);
    o.z = fmaxf(v.z + bias[q + 2], 0.0f);
    o.w = fmaxf(v.w + bias[q + 3], 0.0f);
    *(float4*)(out + (size_t)i * 64 + q) = o;
}

// global_add_pool: hg[batch[i]] += h[i]
__global__ void pool_kernel(const float* __restrict__ h, const int* __restrict__ batch,
                            float* __restrict__ hg, int n) {
    int t = blockIdx.x * blockDim.x + threadIdx.x;
    int i = t >> 4;
    if (i >= n) return;
    int q = (t & 15) * 4;
    int g = batch[i];
    const float4 v = *(const float4*)(h + (size_t)i * 64 + q);
    float* ap = hg + (size_t)g * 64 + q;
    atomicAdd(ap + 0, v.x);
    atomicAdd(ap + 1, v.y);
    atomicAdd(ap + 2, v.z);
    atomicAdd(ap + 3, v.w);
}

// ---------------------------------------------------------------------------
// head: BN -> FC+relu -> BN -> classifier -> log_softmax, one block of 256
// ---------------------------------------------------------------------------
__global__ void head_kernel(const float* __restrict__ hg,
                            const float* __restrict__ fc_gamma, const float* __restrict__ fc_beta,
                            const float* __restrict__ W_fc, const float* __restrict__ b_fc,
                            const float* __restrict__ hid_gamma, const float* __restrict__ hid_beta,
                            const float* __restrict__ W_cls, const float* __restrict__ b_cls,
                            float* __restrict__ out) {
    __shared__ float A[G_BATCH][64];
    __shared__ float Bm[G_BATCH][64];
    __shared__ float sc[64];
    __shared__ float sh[64];
    __shared__ float logits[G_BATCH][OUT_DIM];
    int tid = threadIdx.x;

    for (int idx = tid; idx < G_BATCH * 64; idx += 256) A[idx >> 6][idx & 63] = hg[idx];
    __syncthreads();

    if (tid < 64) {
        float s = 0.0f, q = 0.0f;
        for (int g = 0; g < G_BATCH; g++) { float v = A[g][tid]; s += v; q += v * v; }
        float mu = s * (1.0f / G_BATCH);
        float var = q * (1.0f / G_BATCH) - mu * mu;
        float scv = fc_gamma[tid] * rsqrtf(var + BN_EPS);
        sc[tid] = scv;
        sh[tid] = fc_beta[tid] - mu * scv;
    }
    __syncthreads();

    for (int idx = tid; idx < G_BATCH * 64; idx += 256) {
        int g = idx >> 6, j = idx & 63;
        float acc = b_fc[j];
        for (int k = 0; k < 64; k++) acc += (A[g][k] * sc[k] + sh[k]) * W_fc[k * 64 + j];
        Bm[g][j] = fmaxf(acc, 0.0f);
    }
    __syncthreads();

    if (tid < 64) {
        float s = 0.0f, q = 0.0f;
        for (int g = 0; g < G_BATCH; g++) { float v = Bm[g][tid]; s += v; q += v * v; }
        float mu = s * (1.0f / G_BATCH);
        float var = q * (1.0f / G_BATCH) - mu * mu;
        float scv = hid_gamma[tid] * rsqrtf(var + BN_EPS);
        sc[tid] = scv;
        sh[tid] = hid_beta[tid] - mu * scv;
    }
    __syncthreads();

    for (int idx = tid; idx < G_BATCH * OUT_DIM; idx += 256) {
        int g = idx / OUT_DIM, o = idx % OUT_DIM;
        float acc = b_cls[o];
        for (int k = 0; k < 64; k++) acc += (Bm[g][k] * sc[k] + sh[k]) * W_cls[k * OUT_DIM + o];
        logits[g][o] = acc;
    }
    __syncthreads();

    if (tid < G_BATCH) {
        float mx = -1e30f;
        for (int o = 0; o < OUT_DIM; o++) mx = fmaxf(mx, logits[tid][o]);
        float se = 0.0f;
        for (int o = 0; o < OUT_DIM; o++) se += expf(logits[tid][o] - mx);
        float lse = mx + logf(se);
        for (int o = 0; o < OUT_DIM; o++) out[tid * OUT_DIM + o] = logits[tid][o] - lse;
    }
}

// ---------------------------------------------------------------------------
// host launcher
// ---------------------------------------------------------------------------
extern "C" void kernel_launch(void* const* d_in, const int* in_sizes, int n_in,
                              void* d_out, int out_size, void* d_ws, size_t ws_size,
                              hipStream_t stream) {
    const float* x        = (const float*)d_in[0];
    const int*   ei       = (const int*)d_in[1];
    const int*   batch    = (const int*)d_in[2];
    const float* bnf_g    = (const float*)d_in[3];
    const float* bnf_b    = (const float*)d_in[4];
    const float* W_feat   = (const float*)d_in[5];
    const float* b_feat   = (const float*)d_in[6];
    const float* cbn_g    = (const float*)d_in[7];
    const float* cbn_b    = (const float*)d_in[8];
    const float* conv_W   = (const float*)d_in[9];
    const float* conv_b   = (const float*)d_in[10];
    const float* bnfc_g   = (const float*)d_in[11];
    const float* bnfc_b   = (const float*)d_in[12];
    const float* W_fc     = (const float*)d_in[13];
    const float* b_fc     = (const float*)d_in[14];
    const float* bnh_g    = (const float*)d_in[15];
    const float* bnh_b    = (const float*)d_in[16];
    const float* W_cls    = (const float*)d_in[17];
    const float* b_cls    = (const float*)d_in[18];
    float* out = (float*)d_out;

    const int* src = ei;            // edge_index[0]
    const int* dst = ei + N_EDGES;  // edge_index[1]

    // workspace layout (floats)
    float* ws = (float*)d_ws;
    float* dis    = ws;                       // N
    float* norm   = dis + N_NODES;            // E
    float* buf0   = norm + N_EDGES;           // N*64
    float* buf1   = buf0 + (size_t)N_NODES * 64;
    float* buf2   = buf1 + (size_t)N_NODES * 64;
    float* gsum   = buf2 + (size_t)N_NODES * 64;  // 64
    float* gsumsq = gsum + 64;                     // 64
    float* scale  = gsumsq + 64;                   // 64
    float* shift  = scale + 64;                    // 64
    float* hg     = shift + 64;                    // G*64

    const int TB = 256;
    int gN   = (N_NODES + TB - 1) / TB;
    int gE   = (N_EDGES + TB - 1) / TB;
    int gN16 = (N_NODES * 16 + TB - 1) / TB;  // 16 threads per node (float4 x 4)
    int gE16 = (N_EDGES * 16 + TB - 1) / TB;  // 16 threads per edge
    int gGemm = ((N_NODES / 16) + 7) / 8;     // 8 waves per 256-thread block
    const int STAT_BLOCKS = 256;

    // --- gcn_norm: degree (self loops folded as init=1), rsqrt, edge norms
    fill_f32<<<gN, TB, 0, stream>>>(dis, 1.0f, N_NODES);
    deg_accum<<<gE, TB, 0, stream>>>(src, dis, N_EDGES);
    deg_to_dis<<<gN, TB, 0, stream>>>(dis, N_NODES);
    norm_kernel<<<gE, TB, 0, stream>>>(src, dst, dis, norm, N_EDGES);

    // --- feat block: h = relu(bn(x) @ W_feat + b_feat)
    zero_f32<<<1, 128, 0, stream>>>(gsum, 128);  // gsum + gsumsq contiguous
    bn_stats<<<STAT_BLOCKS, TB, 0, stream>>>(x, gsum, gsumsq, N_NODES);
    bn_finalize<<<1, 64, 0, stream>>>(gsum, gsumsq, bnf_g, bnf_b, scale, shift, N_NODES);
    gemm_bn_wmma<<<gGemm, TB, 0, stream>>>(x, W_feat, scale, shift, b_feat, buf0, N_NODES, 1);

    // --- conv layers
    for (int l = 0; l < L_LAYERS; l++) {
        zero_f32<<<1, 128, 0, stream>>>(gsum, 128);
        bn_stats<<<STAT_BLOCKS, TB, 0, stream>>>(buf0, gsum, gsumsq, N_NODES);
        bn_finalize<<<1, 64, 0, stream>>>(gsum, gsumsq, cbn_g + l * 64, cbn_b + l * 64,
                                          scale, shift, N_NODES);
        gemm_bn_wmma<<<gGemm, TB, 0, stream>>>(buf0, conv_W + (size_t)l * 64 * 64,
                                               scale, shift, nullptr, buf1, N_NODES, 0);
        selfloop_init<<<gN16, TB, 0, stream>>>(buf1, dis, buf2, N_NODES);
        edge_scatter<<<gE16, TB, 0, stream>>>(buf1, src, dst, norm, buf2, N_EDGES);
        bias_relu<<<gN16, TB, 0, stream>>>(buf2, conv_b + l * 64, buf0, N_NODES);
    }

    // --- global_add_pool
    zero_f32<<<(G_BATCH * 64 + TB - 1) / TB, TB, 0, stream>>>(hg, G_BATCH * 64);
    pool_kernel<<<gN16, TB, 0, stream>>>(buf0, batch, hg, N_NODES);

    // --- head
    head_kernel<<<1, 256, 0, stream>>>(hg, bnfc_g, bnfc_b, W_fc, b_fc,
                                       bnh_g, bnh_b, W_cls, b_cls, out);

    (void)in_sizes; (void)n_in; (void)out_size; (void)ws_size;
}